// MambaBlock_47270410060482
// MI455X (gfx1250) — compile-verified
//
#include <hip/hip_runtime.h>
#include <math.h>

#define DIM      1024
#define D_INNER  2048
#define D_STATE  16
#define D_CONV   4
#define DT_RANK  64
#define B_SZ     2
#define LSEQ     2048
#define NTOK     (B_SZ * LSEQ)   // 4096 tokens
#define NCH      16              // scan chunks
#define CLEN     (LSEQ / NCH)    // 128 steps per chunk

typedef float v2f __attribute__((ext_vector_type(2)));
typedef float v8f __attribute__((ext_vector_type(8)));

// ---------------------------------------------------------------------------
// Async global -> LDS (gfx1250), tracked by ASYNCcnt.
// ---------------------------------------------------------------------------
__device__ __forceinline__ void async_ld_b128(const float* gsrc, const float* ldst) {
    const unsigned loff = (unsigned)(uintptr_t)ldst;   // flat addr low 32b = LDS offset
    asm volatile("global_load_async_to_lds_b128 %0, %1, off"
                 :: "v"(loff), "v"(gsrc)
                 : "memory");
}
__device__ __forceinline__ void wait_asynccnt0() {
    asm volatile("s_wait_asynccnt 0x0" ::: "memory");
}

// ---------------------------------------------------------------------------
// LayerNorm over DIM per token row.
// ---------------------------------------------------------------------------
__global__ __launch_bounds__(256)
void ln_kernel(const float* __restrict__ x, const float* __restrict__ gam,
               const float* __restrict__ bet, float* __restrict__ xn) {
    __shared__ float s1[256];
    __shared__ float s2[256];
    const int row = blockIdx.x;
    const float* xr = x + (size_t)row * DIM;
    float v[4];
    float sum = 0.f, sq = 0.f;
#pragma unroll
    for (int i = 0; i < 4; ++i) {
        v[i] = xr[threadIdx.x + 256 * i];
        sum += v[i];
        sq  += v[i] * v[i];
    }
    s1[threadIdx.x] = sum; s2[threadIdx.x] = sq;
    __syncthreads();
    for (int off = 128; off > 0; off >>= 1) {
        if (threadIdx.x < off) {
            s1[threadIdx.x] += s1[threadIdx.x + off];
            s2[threadIdx.x] += s2[threadIdx.x + off];
        }
        __syncthreads();
    }
    const float mean = s1[0] * (1.0f / DIM);
    const float var  = s2[0] * (1.0f / DIM) - mean * mean;
    const float rstd = rsqrtf(var + 1e-5f);
    float* outr = xn + (size_t)row * DIM;
#pragma unroll
    for (int i = 0; i < 4; ++i) {
        const int c = threadIdx.x + 256 * i;
        outr[c] = (v[i] - mean) * rstd * gam[c] + bet[c];
    }
}

// ---------------------------------------------------------------------------
// fp32 WMMA GEMM:  C[M,N] = A[M,K] @ B[K,N]  (+ epilogue)
// 256 threads (8 wave32), 64x64 C tile, double-buffered K chunks of 32 staged
// into LDS with global_load_async_to_lds_b128 (ASYNCcnt). Each wave: 32x16
// region = two v_wmma_f32_16x16x4_f32 accumulators.
// Out-of-range N columns are loaded clamped (garbage only feeds columns that
// are never stored -> safe by column independence of matmul).
// EPI: 0 = store, 1 = softplus(acc + bias[n]), 2 = acc + residual[m,n]
// ---------------------------------------------------------------------------
template <int EPI>
__global__ __launch_bounds__(256)
void gemm_wmma_f32(const float* __restrict__ A, int lda,
                   const float* __restrict__ B, int ldb,
                   float* __restrict__ C, int ldc,
                   int M, int N, int K,
                   const float* __restrict__ E, int lde) {
    __shared__ float As[2][64][36];   // padded rows: 16B aligned, conflict-free
    __shared__ float Bs[2][32][68];

    const int m0   = blockIdx.y * 64;
    const int n0   = blockIdx.x * 64;
    const int t    = threadIdx.x;
    const int wave = t >> 5;
    const int lane = t & 31;
    const int wm   = wave >> 2;    // 0..1 : M offset wm*32
    const int wn   = wave & 3;     // 0..3 : N offset wn*16
    const int kh   = lane >> 4;    // lane half selects K pair
    const int ln16 = lane & 15;

    v8f acc0 = {};
    v8f acc1 = {};

    const int arow = t >> 2;          // 0..63
    const int acol = (t & 3) * 8;     // 0/8/16/24
    const int brow = t >> 3;          // 0..31
    const int bcol = (t & 7) * 8;     // 0..56

    // clamped global source coordinates (clamp keeps reads in-bounds; any
    // wrong data lands only in rows/cols whose stores are guarded off)
    int gm = m0 + arow; if (gm >= M) gm = M - 1;
    int cb0 = n0 + bcol;     if (cb0 + 3 >= N) cb0 = N - 4;
    int cb1 = n0 + bcol + 4; if (cb1 + 3 >= N) cb1 = N - 4;

    // prologue: stage chunk 0 into buffer 0
    {
        const float* asrc = A + (size_t)gm * lda + acol;
        async_ld_b128(asrc,     &As[0][arow][acol]);
        async_ld_b128(asrc + 4, &As[0][arow][acol + 4]);
        const float* brp = B + (size_t)brow * ldb;
        async_ld_b128(brp + cb0, &Bs[0][brow][bcol]);
        async_ld_b128(brp + cb1, &Bs[0][brow][bcol + 4]);
    }

    int buf = 0;
    for (int k0 = 0; k0 < K; k0 += 32) {
        wait_asynccnt0();      // my async writes for current buffer landed
        __syncthreads();       // everyone's landed; prev compute finished

        if (k0 + 32 < K) {     // prefetch next chunk into other buffer
            const int kn = k0 + 32;
            const float* asrc = A + (size_t)gm * lda + kn + acol;
            async_ld_b128(asrc,     &As[buf ^ 1][arow][acol]);
            async_ld_b128(asrc + 4, &As[buf ^ 1][arow][acol + 4]);
            const float* brp = B + (size_t)(kn + brow) * ldb;
            async_ld_b128(brp + cb0, &Bs[buf ^ 1][brow][bcol]);
            async_ld_b128(brp + cb1, &Bs[buf ^ 1][brow][bcol + 4]);
        }

#pragma unroll
        for (int kk = 0; kk < 8; ++kk) {
            const int kb = kk * 4 + kh * 2;
            const int ma = wm * 32 + ln16;
            const int nb = wn * 16 + ln16;
            v2f a0, a1, bf;
            a0.x = As[buf][ma][kb];       a0.y = As[buf][ma][kb + 1];
            a1.x = As[buf][ma + 16][kb];  a1.y = As[buf][ma + 16][kb + 1];
            bf.x = Bs[buf][kb][nb];       bf.y = Bs[buf][kb + 1][nb];
            acc0 = __builtin_amdgcn_wmma_f32_16x16x4_f32(
                false, a0, false, bf, (short)0, acc0, false, false);
            acc1 = __builtin_amdgcn_wmma_f32_16x16x4_f32(
                false, a1, false, bf, (short)0, acc1, false, false);
        }
        buf ^= 1;
    }

    // --- epilogue + store ---
    const int colg = n0 + wn * 16 + ln16;
    if (colg < N) {
#pragma unroll
        for (int i = 0; i < 8; ++i) {
            const int r0 = m0 + wm * 32 + i + 8 * kh;
            const int r1 = r0 + 16;
            float v0 = acc0[i];
            float v1 = acc1[i];
            if constexpr (EPI == 1) {   // + bias, softplus
                v0 += E[colg];
                v1 += E[colg];
                v0 = (v0 > 20.f) ? v0 : log1pf(expf(v0));
                v1 = (v1 > 20.f) ? v1 : log1pf(expf(v1));
            } else if constexpr (EPI == 2) {  // + residual
                if (r0 < M) v0 += E[(size_t)r0 * lde + colg];
                if (r1 < M) v1 += E[(size_t)r1 * lde + colg];
            }
            if (r0 < M) C[(size_t)r0 * ldc + colg] = v0;
            if (r1 < M) C[(size_t)r1 * ldc + colg] = v1;
        }
    }
}

// ---------------------------------------------------------------------------
// Causal depthwise conv (width 4) + SiLU on the x_in half of xz.
// ---------------------------------------------------------------------------
__global__ __launch_bounds__(256)
void conv_silu_kernel(const float* __restrict__ xz,
                      const float* __restrict__ cw,
                      const float* __restrict__ cb,
                      float* __restrict__ xc) {
    const size_t idx = (size_t)blockIdx.x * 256 + threadIdx.x;  // NTOK*D_INNER
    const int d   = (int)(idx % D_INNER);
    const int tok = (int)(idx / D_INNER);
    const int l   = tok % LSEQ;
    const float4 w4 = *(const float4*)(cw + (size_t)d * 4);
    const float wj[4] = {w4.x, w4.y, w4.z, w4.w};
    float acc = cb[d];
#pragma unroll
    for (int j = 0; j < 4; ++j) {
        const int ll = l - 3 + j;
        if (ll >= 0)
            acc += xz[(size_t)(tok - 3 + j) * (2 * D_INNER) + d] * wj[j];
    }
    acc = acc / (1.f + expf(-acc));   // silu
    xc[idx] = acc;
}

// ---------------------------------------------------------------------------
// Chunked parallel selective scan.
// Pass A: per (b, chunk, d): chunk-final state S[16] from h=0, and sum(dt).
// ---------------------------------------------------------------------------
__global__ __launch_bounds__(256)
void scan_chunk_kernel(const float* __restrict__ dt, const float* __restrict__ xc,
                       const float* __restrict__ dbc, const float* __restrict__ A_log,
                       float* __restrict__ Sbuf, float* __restrict__ dtsum) {
    const int gid = blockIdx.x * 256 + threadIdx.x;  // B_SZ*NCH*D_INNER
    const int d   = gid % D_INNER;
    const int tmp = gid / D_INNER;
    const int ch  = tmp % NCH;
    const int b   = tmp / NCH;

    float a[D_STATE], h[D_STATE];
#pragma unroll
    for (int n = 0; n < D_STATE; ++n) {
        a[n] = -expf(A_log[(size_t)d * D_STATE + n]);
        h[n] = 0.f;
    }
    float dts = 0.f;
    const int l0 = ch * CLEN;
    for (int l = l0; l < l0 + CLEN; ++l) {
        const size_t tok = (size_t)b * LSEQ + l;
        const float dtv = dt[tok * D_INNER + d];
        const float xcv = xc[tok * D_INNER + d];
        const float4* Bp = (const float4*)(dbc + tok * 96 + DT_RANK);
        float Bv[D_STATE];
#pragma unroll
        for (int q = 0; q < 4; ++q) {
            const float4 bb = Bp[q];
            Bv[q * 4 + 0] = bb.x; Bv[q * 4 + 1] = bb.y;
            Bv[q * 4 + 2] = bb.z; Bv[q * 4 + 3] = bb.w;
        }
        const float dx = dtv * xcv;
        dts += dtv;
#pragma unroll
        for (int n = 0; n < D_STATE; ++n)
            h[n] = expf(dtv * a[n]) * h[n] + dx * Bv[n];
    }
    float* so = Sbuf + (size_t)gid * D_STATE;
#pragma unroll
    for (int n = 0; n < D_STATE; ++n) so[n] = h[n];
    dtsum[gid] = dts;
}

// ---------------------------------------------------------------------------
// Pass B: per (b, d): sequentially combine the 16 chunks into start states.
// Chunk transfer operator: h_next = S[ch] + exp(a * sum_dt[ch]) * h.
// ---------------------------------------------------------------------------
__global__ __launch_bounds__(256)
void scan_combine_kernel(const float* __restrict__ Sbuf, const float* __restrict__ dtsum,
                         const float* __restrict__ A_log, float* __restrict__ Hstart) {
    const int gid = blockIdx.x * 256 + threadIdx.x;  // B_SZ*D_INNER
    const int b = gid / D_INNER;
    const int d = gid % D_INNER;
    float a[D_STATE], h[D_STATE];
#pragma unroll
    for (int n = 0; n < D_STATE; ++n) {
        a[n] = -expf(A_log[(size_t)d * D_STATE + n]);
        h[n] = 0.f;
    }
    for (int ch = 0; ch < NCH; ++ch) {
        const size_t base = ((size_t)(b * NCH + ch) * D_INNER + d) * D_STATE;
        float* ho = Hstart + base;
#pragma unroll
        for (int n = 0; n < D_STATE; ++n) ho[n] = h[n];
        if (ch < NCH - 1) {
            const float dts = dtsum[(size_t)(b * NCH + ch) * D_INNER + d];
            const float* sp = Sbuf + base;
#pragma unroll
            for (int n = 0; n < D_STATE; ++n)
                h[n] = sp[n] + expf(a[n] * dts) * h[n];
        }
    }
}

// ---------------------------------------------------------------------------
// Pass C: per (b, chunk, d): rerun chunk from its true start state, emit
// y = (scan + D*xc) * silu(z). Writes y over the dt buffer (read-then-write
// of the same element by the same thread -> safe).
// ---------------------------------------------------------------------------
__global__ __launch_bounds__(256)
void scan_final_kernel(const float* dt, const float* __restrict__ xc,
                       const float* __restrict__ xz, const float* __restrict__ dbc,
                       const float* __restrict__ A_log, const float* __restrict__ Dp,
                       const float* __restrict__ Hstart, float* y) {
    const int gid = blockIdx.x * 256 + threadIdx.x;  // B_SZ*NCH*D_INNER
    const int d   = gid % D_INNER;
    const int tmp = gid / D_INNER;
    const int ch  = tmp % NCH;
    const int b   = tmp / NCH;

    float a[D_STATE], h[D_STATE];
    const float* hs = Hstart + (size_t)gid * D_STATE;
#pragma unroll
    for (int n = 0; n < D_STATE; ++n) {
        a[n] = -expf(A_log[(size_t)d * D_STATE + n]);
        h[n] = hs[n];
    }
    const float Dd = Dp[d];
    const int l0 = ch * CLEN;
    for (int l = l0; l < l0 + CLEN; ++l) {
        const size_t tok = (size_t)b * LSEQ + l;
        const float dtv = dt[tok * D_INNER + d];
        const float xcv = xc[tok * D_INNER + d];
        const float zv  = xz[tok * (2 * D_INNER) + D_INNER + d];
        const float4* Bp = (const float4*)(dbc + tok * 96 + DT_RANK);
        const float4* Cp = (const float4*)(dbc + tok * 96 + DT_RANK + D_STATE);
        float Bv[D_STATE], Cv[D_STATE];
#pragma unroll
        for (int q = 0; q < 4; ++q) {
            const float4 bb = Bp[q];
            const float4 cc = Cp[q];
            Bv[q * 4 + 0] = bb.x; Bv[q * 4 + 1] = bb.y;
            Bv[q * 4 + 2] = bb.z; Bv[q * 4 + 3] = bb.w;
            Cv[q * 4 + 0] = cc.x; Cv[q * 4 + 1] = cc.y;
            Cv[q * 4 + 2] = cc.z; Cv[q * 4 + 3] = cc.w;
        }
        const float dx = dtv * xcv;
        float yv = 0.f;
#pragma unroll
        for (int n = 0; n < D_STATE; ++n) {
            h[n] = expf(dtv * a[n]) * h[n] + dx * Bv[n];
            yv += h[n] * Cv[n];
        }
        yv += Dd * xcv;
        const float sz = zv / (1.f + expf(-zv));     // silu(z)
        y[tok * D_INNER + d] = yv * sz;
    }
}

// ---------------------------------------------------------------------------
extern "C" void kernel_launch(void* const* d_in, const int* in_sizes, int n_in,
                              void* d_out, int out_size, void* d_ws, size_t ws_size,
                              hipStream_t stream) {
    (void)in_sizes; (void)n_in; (void)out_size; (void)ws_size;

    const float* x       = (const float*)d_in[0];   // [B,L,DIM]
    const float* ln_g    = (const float*)d_in[1];
    const float* ln_b    = (const float*)d_in[2];
    const float* W_in    = (const float*)d_in[3];   // [DIM, 2*D_INNER]
    const float* conv_w  = (const float*)d_in[4];   // [D_INNER, 4]
    const float* conv_b  = (const float*)d_in[5];
    const float* W_xproj = (const float*)d_in[6];   // [D_INNER, 96]
    const float* W_dt    = (const float*)d_in[7];   // [DT_RANK, D_INNER]
    const float* b_dt    = (const float*)d_in[8];
    const float* A_log   = (const float*)d_in[9];   // [D_INNER, 16]
    const float* Dvec    = (const float*)d_in[10];
    const float* W_out   = (const float*)d_in[11];  // [D_INNER, DIM]
    float* out = (float*)d_out;                     // [B,L,DIM]

    // workspace layout (floats)
    float* xn    = (float*)d_ws;                        // NTOK x DIM
    float* xz    = xn    + (size_t)NTOK * DIM;          // NTOK x 2*D_INNER
    float* xc    = xz    + (size_t)NTOK * 2 * D_INNER;  // NTOK x D_INNER
    float* dbc   = xc    + (size_t)NTOK * D_INNER;      // NTOK x 96
    float* dtb   = dbc   + (size_t)NTOK * 96;           // NTOK x D_INNER (dt, then y)
    float* Sbuf  = dtb   + (size_t)NTOK * D_INNER;      // B*NCH*D_INNER*16
    float* dts   = Sbuf  + (size_t)B_SZ * NCH * D_INNER * D_STATE; // B*NCH*D_INNER
    float* Hst   = dts   + (size_t)B_SZ * NCH * D_INNER;           // B*NCH*D_INNER*16

    // 1) LayerNorm
    ln_kernel<<<NTOK, 256, 0, stream>>>(x, ln_g, ln_b, xn);

    // 2) in_proj: xz = xn @ W_in   [4096 x 4096]
    {
        dim3 grid((2 * D_INNER) / 64, NTOK / 64);
        gemm_wmma_f32<0><<<grid, 256, 0, stream>>>(xn, DIM, W_in, 2 * D_INNER,
                                                   xz, 2 * D_INNER,
                                                   NTOK, 2 * D_INNER, DIM,
                                                   nullptr, 0);
    }

    // 3) causal depthwise conv + SiLU -> xc
    conv_silu_kernel<<<(NTOK * D_INNER) / 256, 256, 0, stream>>>(xz, conv_w, conv_b, xc);

    // 4) x_proj: dbc = xc @ W_xproj   [4096 x 96]
    {
        dim3 grid((96 + 63) / 64, NTOK / 64);
        gemm_wmma_f32<0><<<grid, 256, 0, stream>>>(xc, D_INNER, W_xproj, 96,
                                                   dbc, 96,
                                                   NTOK, 96, D_INNER,
                                                   nullptr, 0);
    }

    // 5) dt = softplus(dbc[:, :64] @ W_dt + b_dt)   [4096 x 2048]
    {
        dim3 grid(D_INNER / 64, NTOK / 64);
        gemm_wmma_f32<1><<<grid, 256, 0, stream>>>(dbc, 96, W_dt, D_INNER,
                                                   dtb, D_INNER,
                                                   NTOK, D_INNER, DT_RANK,
                                                   b_dt, 0);
    }

    // 6) chunked parallel selective scan (+ D*xc + silu(z) gating)
    scan_chunk_kernel<<<(B_SZ * NCH * D_INNER) / 256, 256, 0, stream>>>(
        dtb, xc, dbc, A_log, Sbuf, dts);
    scan_combine_kernel<<<(B_SZ * D_INNER) / 256, 256, 0, stream>>>(
        Sbuf, dts, A_log, Hst);
    scan_final_kernel<<<(B_SZ * NCH * D_INNER) / 256, 256, 0, stream>>>(
        dtb, xc, xz, dbc, A_log, Dvec, Hst, dtb);

    // 7) out = x + y @ W_out   [4096 x 1024]
    {
        dim3 grid(DIM / 64, NTOK / 64);
        gemm_wmma_f32<2><<<grid, 256, 0, stream>>>(dtb, D_INNER, W_out, DIM,
                                                   out, DIM,
                                                   NTOK, DIM, D_INNER,
                                                   x, DIM);
    }
}